// NCutsLoss_28793460752709
// MI455X (gfx1250) — compile-verified
//
#include <hip/hip_runtime.h>

typedef float v2f __attribute__((ext_vector_type(2)));
typedef float v8f __attribute__((ext_vector_type(8)));

#define N_ 8
#define K_ 32
#define P_ (512 * 512)
#define BLOCK_PIX 4096
#define CHUNKS_PER_N (P_ / BLOCK_PIX)   // 64
#define WAVE_PIX (BLOCK_PIX / 8)        // 512 pixels per wave

// workspace layout (floats): acc[N_*K_*8] (S, ws, wm0..2, pad) then qm[N_*4] (Q, M0..2)

__global__ void ncuts_zero_ws(float* ws, int n) {
    int i = blockIdx.x * blockDim.x + threadIdx.x;
    if (i < n) ws[i] = 0.0f;
}

__global__ __launch_bounds__(256) void ncuts_moments(
    const float* __restrict__ seg,   // [N, K, P]
    const float* __restrict__ x,     // [N, C, P]
    float* __restrict__ acc,         // [N, K, 8]
    float* __restrict__ qm)          // [N, 4]
{
    __shared__ float4 tile[BLOCK_PIX];  // {sq, x0, x1, x2} per pixel, 64 KB

    const int b     = blockIdx.x;
    const int n     = b / CHUNKS_PER_N;
    const int chunk = b % CHUNKS_PER_N;
    const long base = (long)chunk * BLOCK_PIX;
    const int  t    = threadIdx.x;
    const int  lane = t & 31;
    const int  wave = t >> 5;

    // ---- Stage x into LDS (coalesced), accumulate Q / M partials ----
    const float4* x0p = (const float4*)(x + ((long)n * 3 + 0) * P_ + base);
    const float4* x1p = (const float4*)(x + ((long)n * 3 + 1) * P_ + base);
    const float4* x2p = (const float4*)(x + ((long)n * 3 + 2) * P_ + base);

    float psq = 0.f, p0 = 0.f, p1 = 0.f, p2 = 0.f;
    #pragma unroll
    for (int j = 0; j < BLOCK_PIX / 4 / 256; ++j) {   // 4 iterations
        int idx = j * 256 + t;                         // float4-group index
        float4 a = x0p[idx];
        float4 c = x1p[idx];
        float4 e = x2p[idx];
        float s0 = a.x * a.x + c.x * c.x + e.x * e.x;
        float s1 = a.y * a.y + c.y * c.y + e.y * e.y;
        float s2 = a.z * a.z + c.z * c.z + e.z * e.z;
        float s3 = a.w * a.w + c.w * c.w + e.w * e.w;
        tile[4 * idx + 0] = make_float4(s0, a.x, c.x, e.x);
        tile[4 * idx + 1] = make_float4(s1, a.y, c.y, e.y);
        tile[4 * idx + 2] = make_float4(s2, a.z, c.z, e.z);
        tile[4 * idx + 3] = make_float4(s3, a.w, c.w, e.w);
        psq += s0 + s1 + s2 + s3;
        p0  += a.x + a.y + a.z + a.w;
        p1  += c.x + c.y + c.z + c.w;
        p2  += e.x + e.y + e.z + e.w;
    }
    // wave32 reduction, one atomic per wave
    #pragma unroll
    for (int off = 16; off > 0; off >>= 1) {
        psq += __shfl_xor(psq, off);
        p0  += __shfl_xor(p0,  off);
        p1  += __shfl_xor(p1,  off);
        p2  += __shfl_xor(p2,  off);
    }
    if (lane == 0) {
        atomicAdd(&qm[n * 4 + 0], psq);
        atomicAdd(&qm[n * 4 + 1], p0);
        atomicAdd(&qm[n * 4 + 2], p1);
        atomicAdd(&qm[n * 4 + 3], p2);
    }
    __syncthreads();

    // ---- WMMA moment accumulation: D[k, col] += seg[16k x 4pix] * B[4pix x 16col] ----
    const int  wbase = wave * WAVE_PIX;
    const int  col   = lane & 15;     // A: M row / B: N column
    const int  half  = lane >> 4;     // selects inner-K (pixel) pair
    const float* pa0 = seg + ((long)n * K_ + col) * P_ + base + wbase + 2 * half;
    const float* pa1 = pa0 + (long)16 * P_;

    // Branchless B-column builder: column col of B is [1, sq, x0, x1, x2, 0...0]
    const float basef = (col == 0) ? 1.0f : 0.0f;
    const float w1    = (col == 1) ? 1.0f : 0.0f;
    const float w2    = (col == 2) ? 1.0f : 0.0f;
    const float w3    = (col == 3) ? 1.0f : 0.0f;
    const float w4    = (col == 4) ? 1.0f : 0.0f;

    v8f d0 = {};
    v8f d1 = {};
    #pragma unroll 2
    for (int it = 0; it < WAVE_PIX / 4; ++it) {
        int q0 = wbase + it * 4 + 2 * half;            // pixel for component 0
        float4 m0 = tile[q0];                          // LDS broadcast (uniform per half)
        float4 m1 = tile[q0 + 1];
        v2f bf;
        bf.x = fmaf(m0.w, w4, fmaf(m0.z, w3, fmaf(m0.y, w2, fmaf(m0.x, w1, basef))));
        bf.y = fmaf(m1.w, w4, fmaf(m1.z, w3, fmaf(m1.y, w2, fmaf(m1.x, w1, basef))));
        v2f a0 = *(const v2f*)(pa0 + it * 4);
        v2f a1 = *(const v2f*)(pa1 + it * 4);
        __builtin_prefetch(pa0 + it * 4 + 512, 0, 3);  // near-cache streaming prefetch
        __builtin_prefetch(pa1 + it * 4 + 512, 0, 3);
        d0 = __builtin_amdgcn_wmma_f32_16x16x4_f32(false, a0, false, bf, (short)0, d0, false, false);
        d1 = __builtin_amdgcn_wmma_f32_16x16x4_f32(false, a1, false, bf, (short)0, d1, false, false);
    }

    // ---- Drain D into global accumulators (cols 0..4 live) ----
    if (col < 5) {
        #pragma unroll
        for (int v = 0; v < 8; ++v) {
            int km = v + 8 * half;   // M index within 16-row group
            atomicAdd(&acc[((long)n * K_ + km) * 8 + col],        d0[v]);
            atomicAdd(&acc[((long)n * K_ + 16 + km) * 8 + col],   d1[v]);
        }
    }
}

__global__ void ncuts_finalize(const float* __restrict__ acc,
                               const float* __restrict__ qm,
                               float* __restrict__ out)
{
    __shared__ float red[256];
    int t = threadIdx.x;              // 256 threads == N_*K_ pairs
    int n = t >> 5, k = t & 31;
    const float* a = &acc[((long)n * K_ + k) * 8];
    float S  = a[0], ws = a[1];
    float w0 = a[2], w1 = a[3], w2 = a[4];
    float Q  = qm[n * 4 + 0];
    float M0 = qm[n * 4 + 1], M1 = qm[n * 4 + 2], M2 = qm[n * 4 + 3];

    float assocA = 2.0f * S * ws - 2.0f * (w0 * w0 + w1 * w1 + w2 * w2);
    float assocV = (float)P_ * ws + S * Q - 2.0f * (w0 * M0 + w1 * M1 + w2 * M2);
    red[t] = assocA / assocV;
    __syncthreads();
    #pragma unroll
    for (int s = 128; s > 0; s >>= 1) {
        if (t < s) red[t] += red[t + s];
        __syncthreads();
    }
    if (t == 0) out[0] = (float)(N_ * K_) - red[0];
}

extern "C" void kernel_launch(void* const* d_in, const int* in_sizes, int n_in,
                              void* d_out, int out_size, void* d_ws, size_t ws_size,
                              hipStream_t stream) {
    const float* seg = (const float*)d_in[0];
    const float* x   = (const float*)d_in[1];
    float* acc = (float*)d_ws;            // N_*K_*8 = 2048 floats
    float* qm  = acc + N_ * K_ * 8;       // N_*4   = 32 floats

    int ztotal = N_ * K_ * 8 + N_ * 4;
    ncuts_zero_ws<<<(ztotal + 255) / 256, 256, 0, stream>>>(acc, ztotal);
    ncuts_moments<<<N_ * CHUNKS_PER_N, 256, 0, stream>>>(seg, x, acc, qm);
    ncuts_finalize<<<1, 256, 0, stream>>>(acc, qm, (float*)d_out);
}